// EmerGNN_71528385347856
// MI455X (gfx1250) — compile-verified
//
#include <hip/hip_runtime.h>
#include <hip/hip_bf16.h>

// ---------------- problem constants (match reference) ----------------
#define N_ENT   30000
#define ALL_REL 221
#define NDIM    32
#define NLAY    2
#define EVALREL 86
#define BB      16
#define BD      (BB * NDIM)          // 512 floats per entity (B,d) slab

typedef float v2f __attribute__((ext_vector_type(2)));
typedef float v8f __attribute__((ext_vector_type(8)));

// Guaranteed hardware f32 atomic add (non-returning, STOREcnt-tracked).
// gfx1250 ISA: GLOBAL_ATOMIC_ADD_F32 (opcode 86). "v" on a 64-bit pointer
// allocates a VGPR pair and prints v[n:n+1].
__device__ __forceinline__ void atomic_add_f32_hw(float* addr, float v) {
    asm volatile("global_atomic_add_f32 %0, %1, off"
                 :
                 : "v"(addr), "v"(v)
                 : "memory");
}

// ---------------- zero fill (float4) ----------------
__global__ void eg_zero_f4(float4* __restrict__ p, int n4) {
    int i = blockIdx.x * blockDim.x + threadIdx.x;
    if (i < n4) p[i] = make_float4(0.f, 0.f, 0.f, 0.f);
}

// ---------------- hid[idx[b], b, :] = ent_emb[idx[b], :] ----------------
__global__ void eg_set_init(float* __restrict__ hid, const int* __restrict__ idx,
                            const float* __restrict__ ent_emb) {
    int t = threadIdx.x;          // 512 threads
    int b = t >> 5, k = t & 31;
    int e = idx[b];
    hid[(size_t)e * BD + b * NDIM + k] = ent_emb[(size_t)e * NDIM + k];
}

// ---------------- hrel[l,b,j] = relu(ht[b,:] @ Wrel[l] + brel[l])  (L*B*5=160) ----
__global__ void eg_hrel(const float* __restrict__ ent_emb, const int* __restrict__ head,
                        const int* __restrict__ tail, const float* __restrict__ Wrel,
                        const float* __restrict__ brel, float* __restrict__ hrel) {
    int t = threadIdx.x;
    if (t >= NLAY * BB * 5) return;
    int l = t / (BB * 5), rem = t % (BB * 5), b = rem / 5, j = rem % 5;
    const float* hrow = ent_emb + (size_t)head[b] * NDIM;
    const float* trow = ent_emb + (size_t)tail[b] * NDIM;
    float s = brel[l * 5 + j];
    #pragma unroll 8
    for (int i = 0; i < NDIM; ++i) s += hrow[i] * Wrel[(l * 64 + i) * 5 + j];
    #pragma unroll 8
    for (int i = 0; i < NDIM; ++i) s += trow[i] * Wrel[(l * 64 + 32 + i) * 5 + j];
    hrel[t] = s > 0.f ? s : 0.f;
}

// ---------- relsc[l,r,b,k] = sigmoid(hrel[l,b]@Watt[l,:,r]+batt[l,r]) * rel_emb[l,r,k] ----
__global__ void eg_relsc(const float* __restrict__ hrel, const float* __restrict__ Watt,
                         const float* __restrict__ batt, const float* __restrict__ rel_embs,
                         float* __restrict__ relsc) {
    int idx = blockIdx.x * blockDim.x + threadIdx.x;
    if (idx >= NLAY * ALL_REL * BD) return;
    int l   = idx / (ALL_REL * BD);
    int rem = idx % (ALL_REL * BD);
    int r   = rem / BD;
    int b   = (rem % BD) >> 5;
    int k   = rem & 31;
    float x = batt[l * ALL_REL + r];
    const float* h = hrel + (l * BB + b) * 5;
    #pragma unroll
    for (int j = 0; j < 5; ++j) x += h[j] * Watt[(l * 5 + j) * ALL_REL + r];
    float s = 1.0f / (1.0f + __expf(-x));
    relsc[idx] = s * rel_embs[(l * ALL_REL + r) * NDIM + k];
}

// ---------------- edge scatter: hidOut[et] += ew * relsc[er] * hidIn[eh] ----------------
// one float4 (of the 512-float (B,d) slab) per thread; 128 threads per edge
__global__ __launch_bounds__(256) void eg_scatter(
    const float* __restrict__ hidIn, float* __restrict__ hidOut,
    const float* __restrict__ relsc_l, const int* __restrict__ eh,
    const int* __restrict__ et, const int* __restrict__ er,
    const float* __restrict__ ew, int nE) {
    long idx = (long)blockIdx.x * blockDim.x + threadIdx.x;
    if (idx >= (long)nE * (BD / 4)) return;
    int e  = (int)(idx >> 7);
    int t4 = (int)(idx & 127);
    int h = eh[e], t = et[e], r = er[e];
    float w = ew[e];
    float4 hv = ((const float4*)(hidIn   + (size_t)h * BD))[t4];
    float4 rv = ((const float4*)(relsc_l + (size_t)r * BD))[t4];
    float* dst = hidOut + (size_t)t * BD + t4 * 4;
    atomic_add_f32_hw(dst + 0, hv.x * rv.x * w);
    atomic_add_f32_hw(dst + 1, hv.y * rv.y * w);
    atomic_add_f32_hw(dst + 2, hv.z * rv.z * w);
    atomic_add_f32_hw(dst + 3, hv.w * rv.w * w);
}

// ---------------- per-entity (16x32)@(32x32)+bias, ReLU via V_WMMA_F32_16X16X4_F32 ------
// one entity per wave32; 8 waves per 256-thread block; N_ENT/8 = 3750 blocks exactly.
// A (16x4 f32): lane<16 -> M=lane, VGPR0/1 = K0/K1 ; lane>=16 -> M=lane-16, K2/K3
// B (4x16 f32): lane<16 -> N=lane, VGPR0/1 = K0/K1 ; lane>=16 -> K2/K3
// C/D (16x16):  lane<16 -> N=lane, VGPR i = M=i   ; lane>=16 -> M=i+8
__global__ __launch_bounds__(256) void eg_linear_relu_wmma(
    const float* __restrict__ hidIn, float* __restrict__ hidOut,
    const float* __restrict__ W,      // (32,32) row-major for this layer
    const float* __restrict__ bias) { // (32,)
    int wave = threadIdx.x >> 5;
    int lane = threadIdx.x & 31;
    int ent  = blockIdx.x * 8 + wave;   // always < N_ENT (3750*8 == 30000)
    int half = lane >> 4;               // K-pair select
    int lm   = lane & 15;

    const float* Arow = hidIn + (size_t)ent * BD + lm * NDIM + 2 * half;
    float bl0 = bias[lm];
    float bl1 = bias[16 + lm];
    v8f acc0, acc1;
    #pragma unroll
    for (int i = 0; i < 8; ++i) { acc0[i] = bl0; acc1[i] = bl1; }

    #pragma unroll
    for (int kc = 0; kc < 8; ++kc) {
        v2f a;
        a.x = Arow[4 * kc];
        a.y = Arow[4 * kc + 1];
        int krow = 4 * kc + 2 * half;
        v2f b0, b1;
        b0.x = W[krow * NDIM + lm];            // N-tile 0
        b0.y = W[(krow + 1) * NDIM + lm];
        b1.x = W[krow * NDIM + 16 + lm];       // N-tile 1
        b1.y = W[(krow + 1) * NDIM + 16 + lm];
        acc0 = __builtin_amdgcn_wmma_f32_16x16x4_f32(false, a, false, b0, (short)0, acc0,
                                                     false, false);
        acc1 = __builtin_amdgcn_wmma_f32_16x16x4_f32(false, a, false, b1, (short)0, acc1,
                                                     false, false);
    }

    float* Drow = hidOut + (size_t)ent * BD + lm;
    #pragma unroll
    for (int i = 0; i < 8; ++i) {
        int m = i + 8 * half;
        float v0 = acc0[i]; v0 = v0 > 0.f ? v0 : 0.f;
        float v1 = acc1[i]; v1 = v1 > 0.f ? v1 : 0.f;
        Drow[m * NDIM]      = v0;
        Drow[m * NDIM + 16] = v1;
    }
}

// ---------------- res[b,k] = hid[idx[b], b, k] ----------------
__global__ void eg_gather(const float* __restrict__ hid, const int* __restrict__ idx,
                          float* __restrict__ res) {
    int t = threadIdx.x;  // 512
    int b = t >> 5, k = t & 31;
    res[t] = hid[(size_t)idx[b] * BD + b * NDIM + k];
}

// ---------------- out[b,r] = concat(he,te,hh,th)[b] @ Wr[:,r] + br[r] ----------------
__global__ void eg_final(const float* __restrict__ ent_emb, const int* __restrict__ head,
                         const int* __restrict__ tail, const float* __restrict__ head_hid,
                         const float* __restrict__ tail_hid, const float* __restrict__ Wr,
                         const float* __restrict__ br, float* __restrict__ out) {
    int idx = blockIdx.x * blockDim.x + threadIdx.x;
    if (idx >= BB * EVALREL) return;
    int b = idx / EVALREL, r = idx % EVALREL;
    const float* he = ent_emb + (size_t)head[b] * NDIM;
    const float* te = ent_emb + (size_t)tail[b] * NDIM;
    float s = br[r];
    #pragma unroll 8
    for (int i = 0; i < NDIM; ++i) s += he[i] * Wr[i * EVALREL + r];
    #pragma unroll 8
    for (int i = 0; i < NDIM; ++i) s += te[i] * Wr[(NDIM + i) * EVALREL + r];
    #pragma unroll 8
    for (int i = 0; i < NDIM; ++i) s += head_hid[b * NDIM + i] * Wr[(2 * NDIM + i) * EVALREL + r];
    #pragma unroll 8
    for (int i = 0; i < NDIM; ++i) s += tail_hid[b * NDIM + i] * Wr[(3 * NDIM + i) * EVALREL + r];
    out[idx] = s;
}

extern "C" void kernel_launch(void* const* d_in, const int* in_sizes, int n_in,
                              void* d_out, int out_size, void* d_ws, size_t ws_size,
                              hipStream_t stream) {
    const int*   head     = (const int*)d_in[0];
    const int*   tail     = (const int*)d_in[1];
    const int*   eh       = (const int*)d_in[2];
    const int*   et       = (const int*)d_in[3];
    const int*   er       = (const int*)d_in[4];
    const float* ew       = (const float*)d_in[5];
    const float* ent_emb  = (const float*)d_in[6];
    const float* rel_embs = (const float*)d_in[7];
    const float* Wlin     = (const float*)d_in[8];
    const float* blin     = (const float*)d_in[9];
    const float* Wrel     = (const float*)d_in[10];
    const float* brel     = (const float*)d_in[11];
    const float* Watt     = (const float*)d_in[12];
    const float* batt     = (const float*)d_in[13];
    const float* Wr       = (const float*)d_in[14];
    const float* br       = (const float*)d_in[15];
    float*       out      = (float*)d_out;
    const int    nE       = in_sizes[2];

    // workspace layout
    float* hidA     = (float*)d_ws;                          // N_ENT*512 floats
    float* hidB     = hidA + (size_t)N_ENT * BD;             // N_ENT*512 floats
    float* relsc    = hidB + (size_t)N_ENT * BD;             // L*221*512 floats
    float* hrel     = relsc + (size_t)NLAY * ALL_REL * BD;   // 160 floats
    float* tail_hid = hrel + 256;                            // 512 floats
    float* head_hid = tail_hid + BD;                         // 512 floats

    const int hid_n4    = (N_ENT * BD) / 4;
    const int zeroGrid  = (hid_n4 + 255) / 256;
    const int relscTot  = NLAY * ALL_REL * BD;
    const long scatTot  = (long)nE * (BD / 4);
    const int scatGrid  = (int)((scatTot + 255) / 256);

    // attention precompute (shared by both directions)
    eg_hrel<<<1, 256, 0, stream>>>(ent_emb, head, tail, Wrel, brel, hrel);
    eg_relsc<<<(relscTot + 255) / 256, 256, 0, stream>>>(hrel, Watt, batt, rel_embs, relsc);

    for (int dir = 0; dir < 2; ++dir) {
        const int* initIdx = (dir == 0) ? head : tail;
        const int* readIdx = (dir == 0) ? tail : head;
        float*     res     = (dir == 0) ? tail_hid : head_hid;

        eg_zero_f4<<<zeroGrid, 256, 0, stream>>>((float4*)hidA, hid_n4);
        eg_set_init<<<1, 512, 0, stream>>>(hidA, initIdx, ent_emb);

        for (int l = 0; l < NLAY; ++l) {
            eg_zero_f4<<<zeroGrid, 256, 0, stream>>>((float4*)hidB, hid_n4);
            eg_scatter<<<scatGrid, 256, 0, stream>>>(hidA, hidB, relsc + (size_t)l * ALL_REL * BD,
                                                     eh, et, er, ew, nE);
            eg_linear_relu_wmma<<<N_ENT / 8, 256, 0, stream>>>(hidB, hidA,
                                                               Wlin + (size_t)l * NDIM * NDIM,
                                                               blin + (size_t)l * NDIM);
        }
        eg_gather<<<1, 512, 0, stream>>>(hidA, readIdx, res);
    }

    eg_final<<<(BB * EVALREL + 255) / 256, 256, 0, stream>>>(ent_emb, head, tail, head_hid,
                                                             tail_hid, Wr, br, out);
}